// HeteLinear_71116068487913
// MI455X (gfx1250) — compile-verified
//
#include <hip/hip_runtime.h>
#include <hip/hip_bf16.h>

typedef float v2f __attribute__((ext_vector_type(2)));
typedef float v8f __attribute__((ext_vector_type(8)));

#define NTYPES 8
#define KDIM   128
#define ODIM   128
#define MTILE  64          // rows per block (4 x 16-row WMMA tiles)

// ---------------- Pass 1: per-type histogram (block-aggregated) --------------
__global__ __launch_bounds__(256) void tl_hist(const int* __restrict__ xt,
                                               int* __restrict__ count, int n) {
    __shared__ int h[NTYPES];
    if (threadIdx.x < NTYPES) h[threadIdx.x] = 0;
    __syncthreads();
    int i = blockIdx.x * blockDim.x + threadIdx.x;
    if (i < n) atomicAdd(&h[xt[i]], 1);
    __syncthreads();
    if (threadIdx.x < NTYPES) atomicAdd(&count[threadIdx.x], h[threadIdx.x]);
}

// ---------------- Pass 2: tiny exclusive scan over 8 bins --------------------
__global__ void tl_scan(const int* __restrict__ count,
                        int* __restrict__ offsets, int* __restrict__ cursor) {
    if (threadIdx.x == 0) {
        int s = 0;
        for (int t = 0; t < NTYPES; ++t) {
            offsets[t] = s; cursor[t] = s; s += count[t];
        }
        offsets[NTYPES] = s;
    }
}

// ---------------- Pass 3: scatter row ids into type-sorted order -------------
__global__ __launch_bounds__(256) void tl_scatter(const int* __restrict__ xt,
                                                  int* __restrict__ cursor,
                                                  int* __restrict__ perm, int n) {
    __shared__ int cnt[NTYPES];
    __shared__ int base[NTYPES];
    if (threadIdx.x < NTYPES) cnt[threadIdx.x] = 0;
    __syncthreads();
    int i = blockIdx.x * blockDim.x + threadIdx.x;
    int t = 0, lr = 0;
    if (i < n) { t = xt[i]; lr = atomicAdd(&cnt[t], 1); }
    __syncthreads();
    if (threadIdx.x < NTYPES) {
        int c = cnt[threadIdx.x];
        base[threadIdx.x] = c ? atomicAdd(&cursor[threadIdx.x], c) : 0;
    }
    __syncthreads();
    if (i < n) perm[base[t] + lr] = i;
}

// ---------------- Pass 4: grouped GEMM over permuted rows, f32 WMMA ----------
// Block = 256 threads = 8 wave32; one block per 64 permuted rows.
// Wave w owns output columns [16w, 16w+16) and accumulates FOUR stacked
// 16x16 D tiles, reusing each B fragment for 4 x v_wmma_f32_16x16x4_f32.
#define XS_STRIDE 132   // pad 128 -> 132 dwords: conflict-free A-frag ds reads

__global__ __launch_bounds__(256) void tl_gemm(const float* __restrict__ x,
                                               const int*   __restrict__ xt,
                                               const float* __restrict__ W,
                                               const float* __restrict__ b,
                                               const int*   __restrict__ perm,
                                               float*       __restrict__ out) {
    __shared__ float Xs[MTILE][XS_STRIDE];
    __shared__ int   permS[MTILE];
    __shared__ int   typeS[MTILE];

    const int tid   = threadIdx.x;
    const int tile0 = blockIdx.x * MTILE;

    if (tid < MTILE) {
        int r = perm[tile0 + tid];
        permS[tid] = r;
        typeS[tid] = xt[r];
    }
    __syncthreads();

    // Gather the 64x128 X tile into LDS: 4 threads/row, 8 x float4 each.
    {
        int row = tid >> 2;
        int c0  = (tid & 3) * 32;
        const float4* src = (const float4*)(x + (size_t)permS[row] * KDIM + c0);
        float4* dst = (float4*)&Xs[row][c0];
#pragma unroll
        for (int j = 0; j < 8; ++j) dst[j] = src[j];
    }
    __syncthreads();

    const int lane    = tid & 31;
    const int wave    = tid >> 5;          // 0..7 -> 16-col slice of output
    const int colbase = wave * 16;
    const int lane15  = lane & 15;
    const int khalf   = (lane >> 4) * 2;   // 0 for lanes 0-15, 2 for 16-31

    // Sorted order => types within the 64-row tile form a contiguous range.
    const int tmin = typeS[0];
    const int tmax = typeS[MTILE - 1];

    const float* Xr0 = &Xs[lane15 +  0][khalf];
    const float* Xr1 = &Xs[lane15 + 16][khalf];
    const float* Xr2 = &Xs[lane15 + 32][khalf];
    const float* Xr3 = &Xs[lane15 + 48][khalf];

    v8f acc0 = {}, acc1 = {}, acc2 = {}, acc3 = {};

    if (tmin == tmax) {
        // Fast path (~99.8% of tiles): single type, B fragment reused 4x.
        const float* Wt = W + (size_t)tmin * KDIM * ODIM + colbase + lane15;
#pragma unroll 4
        for (int k = 0; k < KDIM; k += 4) {
            v2f bb;
            bb.x = Wt[(size_t)(k + khalf) * ODIM];      // B: K=k+khalf
            bb.y = Wt[(size_t)(k + khalf + 1) * ODIM];  //    K=k+khalf+1
            v2f a0, a1, a2, a3;
            a0.x = Xr0[k]; a0.y = Xr0[k + 1];
            a1.x = Xr1[k]; a1.y = Xr1[k + 1];
            a2.x = Xr2[k]; a2.y = Xr2[k + 1];
            a3.x = Xr3[k]; a3.y = Xr3[k + 1];
            acc0 = __builtin_amdgcn_wmma_f32_16x16x4_f32(
                false, a0, false, bb, (short)0, acc0, false, false);
            acc1 = __builtin_amdgcn_wmma_f32_16x16x4_f32(
                false, a1, false, bb, (short)0, acc1, false, false);
            acc2 = __builtin_amdgcn_wmma_f32_16x16x4_f32(
                false, a2, false, bb, (short)0, acc2, false, false);
            acc3 = __builtin_amdgcn_wmma_f32_16x16x4_f32(
                false, a3, false, bb, (short)0, acc3, false, false);
        }
    } else {
        // Boundary tile: accumulate each type in range, masking A rows.
        const int ty0 = typeS[lane15 +  0];
        const int ty1 = typeS[lane15 + 16];
        const int ty2 = typeS[lane15 + 32];
        const int ty3 = typeS[lane15 + 48];
        for (int t = tmin; t <= tmax; ++t) {
            const float s0 = (ty0 == t) ? 1.0f : 0.0f;
            const float s1 = (ty1 == t) ? 1.0f : 0.0f;
            const float s2 = (ty2 == t) ? 1.0f : 0.0f;
            const float s3 = (ty3 == t) ? 1.0f : 0.0f;
            const float* Wt = W + (size_t)t * KDIM * ODIM + colbase + lane15;
#pragma unroll 2
            for (int k = 0; k < KDIM; k += 4) {
                v2f bb;
                bb.x = Wt[(size_t)(k + khalf) * ODIM];
                bb.y = Wt[(size_t)(k + khalf + 1) * ODIM];
                v2f a0, a1, a2, a3;
                a0.x = Xr0[k] * s0; a0.y = Xr0[k + 1] * s0;
                a1.x = Xr1[k] * s1; a1.y = Xr1[k + 1] * s1;
                a2.x = Xr2[k] * s2; a2.y = Xr2[k + 1] * s2;
                a3.x = Xr3[k] * s3; a3.y = Xr3[k + 1] * s3;
                acc0 = __builtin_amdgcn_wmma_f32_16x16x4_f32(
                    false, a0, false, bb, (short)0, acc0, false, false);
                acc1 = __builtin_amdgcn_wmma_f32_16x16x4_f32(
                    false, a1, false, bb, (short)0, acc1, false, false);
                acc2 = __builtin_amdgcn_wmma_f32_16x16x4_f32(
                    false, a2, false, bb, (short)0, acc2, false, false);
                acc3 = __builtin_amdgcn_wmma_f32_16x16x4_f32(
                    false, a3, false, bb, (short)0, acc3, false, false);
            }
        }
    }

    // D layout: VGPR v, lanes 0-15 -> D[M=v][N=lane], lanes 16-31 -> D[M=v+8].
    const int ncol  = colbase + lane15;
    const int mbase = (lane >> 4) * 8;

#pragma unroll
    for (int v = 0; v < 8; ++v) {
        int m = mbase + v;
        out[(size_t)permS[m] * ODIM + ncol] =
            acc0[v] + b[(size_t)typeS[m] * ODIM + ncol];
    }
#pragma unroll
    for (int v = 0; v < 8; ++v) {
        int m = 16 + mbase + v;
        out[(size_t)permS[m] * ODIM + ncol] =
            acc1[v] + b[(size_t)typeS[m] * ODIM + ncol];
    }
#pragma unroll
    for (int v = 0; v < 8; ++v) {
        int m = 32 + mbase + v;
        out[(size_t)permS[m] * ODIM + ncol] =
            acc2[v] + b[(size_t)typeS[m] * ODIM + ncol];
    }
#pragma unroll
    for (int v = 0; v < 8; ++v) {
        int m = 48 + mbase + v;
        out[(size_t)permS[m] * ODIM + ncol] =
            acc3[v] + b[(size_t)typeS[m] * ODIM + ncol];
    }
}

extern "C" void kernel_launch(void* const* d_in, const int* in_sizes, int n_in,
                              void* d_out, int out_size, void* d_ws, size_t ws_size,
                              hipStream_t stream) {
    const float* x  = (const float*)d_in[0];
    const int*   xt = (const int*)  d_in[1];
    const float* W  = (const float*)d_in[2];
    const float* b  = (const float*)d_in[3];
    float*       out = (float*)d_out;

    const int N = in_sizes[1];   // number of rows (== x_type count)

    int* ws      = (int*)d_ws;
    int* count   = ws;           // [8]
    int* offsets = ws + 8;       // [9]
    int* cursor  = ws + 20;      // [8]
    int* perm    = ws + 32;      // [N]

    hipMemsetAsync(count, 0, NTYPES * sizeof(int), stream);

    tl_hist   <<<(N + 255) / 256, 256, 0, stream>>>(xt, count, N);
    tl_scan   <<<1, 32, 0, stream>>>(count, offsets, cursor);
    tl_scatter<<<(N + 255) / 256, 256, 0, stream>>>(xt, cursor, perm, N);
    tl_gemm   <<<N / MTILE, 256, 0, stream>>>(x, xt, W, b, perm, out);
}